// KNN_86071144612542
// MI455X (gfx1250) — compile-verified
//
#include <hip/hip_runtime.h>
#include <hip/hip_bf16.h>
#include <float.h>
#include <math.h>

typedef __attribute__((ext_vector_type(2))) float v2f;
typedef __attribute__((ext_vector_type(8))) float v8f;

#define KNN_K 16
#define KNN_EPS 1e-10f
#define NC 1024               // coords chunk staged in LDS (points), NC*12B contiguous
#define BLOCK_THREADS 128     // 4 waves (wave32)
#define WAVES_PER_BLOCK 4
#define ROWS_PER_WAVE 32      // centers per wave (two 16-row WMMA tiles)
#define ROWS_PER_BLOCK (WAVES_PER_BLOCK * ROWS_PER_WAVE)
#define RAW_B128_PER_THREAD ((NC * 12) / 16 / BLOCK_THREADS)   // 6

__device__ __forceinline__ void async_copy_b128(unsigned lds_addr, unsigned long long gaddr) {
    // GLOBAL_LOAD_ASYNC_TO_LDS_B128 (GV mode): copies 16 bytes from global address vaddr to LDS address vdst; tracked by ASYNCcnt
    asm volatile("global_load_async_to_lds_b128 %0, %1, off"
                 :: "v"(lds_addr), "v"(gaddr) : "memory");
}
__device__ __forceinline__ void wait_async0() {
    asm volatile("s_wait_asynccnt 0" ::: "memory");
}

__global__ __launch_bounds__(BLOCK_THREADS)
void knn_wmma_kernel(const float* __restrict__ coords,     // (B, N, 3)
                     const float* __restrict__ centers,    // (B, M, 3)
                     float* __restrict__ out_dist,         // (B, M, K)
                     int*   __restrict__ out_idx,          // (B, M, K)
                     int N, int M)
{
    __shared__ float  s_raw[NC * 3];                           // raw async-staged xyz
    __shared__ float4 s_pts[NC];                               // x, y, z, |p|^2
    __shared__ float  s_dist[WAVES_PER_BLOCK][ROWS_PER_WAVE][16];

    const int tid  = threadIdx.x;
    const int lane = tid & 31;                 // wave32
    const int wave = tid >> 5;

    const int blocks_per_batch = M / ROWS_PER_BLOCK;
    const int b        = blockIdx.x / blocks_per_batch;
    const int m0_block = (blockIdx.x % blocks_per_batch) * ROWS_PER_BLOCK;
    const int m0       = m0_block + wave * ROWS_PER_WAVE;

    const float* cb = centers + (size_t)b * M * 3;
    const float* pb = coords  + (size_t)b * N * 3;
    const unsigned long long pb_addr = (unsigned long long)(uintptr_t)pb;
    // Low 32 bits of a generic LDS pointer == LDS byte offset (aperture mapping)
    const unsigned raw_base = (unsigned)(uintptr_t)&s_raw[0];

    // ---- A fragments: 16x4 f32, lanes 0-15 hold (K=0,K=1), lanes 16-31 (K=2,K=3)
    // A row m = (-2cx, -2cy, -2cz, 1)  so  A x B(col n = (x,y,z,|p|^2)) = |p|^2 - 2 c.p
    const int  ml     = lane & 15;
    const bool loHalf = (lane < 16);
    const int  mA0 = m0 + ml;
    const int  mA1 = m0 + 16 + ml;
    const float c0x = cb[mA0 * 3 + 0], c0y = cb[mA0 * 3 + 1], c0z = cb[mA0 * 3 + 2];
    const float c1x = cb[mA1 * 3 + 0], c1y = cb[mA1 * 3 + 1], c1z = cb[mA1 * 3 + 2];
    v2f a0, a1;
    a0.x = loHalf ? (-2.0f * c0x) : (-2.0f * c0z);
    a0.y = loHalf ? (-2.0f * c0y) : 1.0f;
    a1.x = loHalf ? (-2.0f * c1x) : (-2.0f * c1z);
    a1.y = loHalf ? (-2.0f * c1y) : 1.0f;

    // |c|^2 of this lane's own row (added back only at finalize; order-invariant)
    const int   m_own = m0 + lane;
    const float ox = cb[m_own * 3 + 0], oy = cb[m_own * 3 + 1], oz = cb[m_own * 3 + 2];
    const float csq = ox * ox + oy * oy + oz * oz;

    // Register-resident sorted top-K (ascending)
    float kd[KNN_K];
    int   ki[KNN_K];
#pragma unroll
    for (int j = 0; j < KNN_K; ++j) { kd[j] = FLT_MAX; ki[j] = 0; }

    // Prologue: async-stage chunk 0 (contiguous NC*12 bytes -> s_raw)
#pragma unroll
    for (int i = 0; i < RAW_B128_PER_THREAD; ++i) {
        const unsigned off = (unsigned)(tid * RAW_B128_PER_THREAD + i) * 16u;
        async_copy_b128(raw_base + off, pb_addr + off);
    }

    for (int chunk = 0; chunk < N; chunk += NC) {
        wait_async0();          // this wave's async stores to LDS complete
        __syncthreads();        // all waves' raw data visible

        // Pack pass: s_raw (xyz) -> s_pts (x,y,z,|p|^2)
        for (int p = tid; p < NC; p += BLOCK_THREADS) {
            const float x = s_raw[p * 3 + 0];
            const float y = s_raw[p * 3 + 1];
            const float z = s_raw[p * 3 + 2];
            s_pts[p] = float4{x, y, z, x * x + y * y + z * z};
        }
        __syncthreads();        // s_pts ready; s_raw fully consumed

        // Overlap: async-stage the NEXT chunk into s_raw while computing this one
        if (chunk + NC < N) {
            const unsigned long long gbase = pb_addr + (unsigned long long)(chunk + NC) * 12u;
#pragma unroll
            for (int i = 0; i < RAW_B128_PER_THREAD; ++i) {
                const unsigned off = (unsigned)(tid * RAW_B128_PER_THREAD + i) * 16u;
                async_copy_b128(raw_base + off, gbase + off);
            }
        }
        // Warm L2 for chunk+2 (global_prefetch_b8)
        if (chunk + 2 * NC < N) {
            __builtin_prefetch(&pb[(size_t)(chunk + 2 * NC) * 3 + (size_t)tid * 24], 0, 1);
        }

        for (int nb = 0; nb < NC; nb += 16) {
            // ---- B fragment: 4x16 f32, mirrors A layout across half-waves
            const float4 p4 = s_pts[nb + ml];
            v2f bf;
            bf.x = loHalf ? p4.x : p4.z;
            bf.y = loHalf ? p4.y : p4.w;

            v8f acc0 = {};
            v8f acc1 = {};
            acc0 = __builtin_amdgcn_wmma_f32_16x16x4_f32(
                false, a0, false, bf, (short)0, acc0, false, false);
            acc1 = __builtin_amdgcn_wmma_f32_16x16x4_f32(
                false, a1, false, bf, (short)0, acc1, false, false);

            // Transpose D tiles through LDS: VGPR r -> row r (+8 for hi half), col = ml
            const int rbase = loHalf ? 0 : 8;
#pragma unroll
            for (int r = 0; r < 8; ++r) {
                s_dist[wave][rbase + r][ml]      = acc0[r];
                s_dist[wave][16 + rbase + r][ml] = acc1[r];
            }
            // Same-wave LDS RAW: DS pipe is in-order per wave; compiler inserts dscnt wait.

            // Each lane owns one center row: 16 candidates as 4x float4 LDS loads
#pragma unroll
            for (int q = 0; q < 4; ++q) {
                const float4 d4 = *(const float4*)&s_dist[wave][lane][q * 4];
                const float dv[4] = {d4.x, d4.y, d4.z, d4.w};
#pragma unroll
                for (int u = 0; u < 4; ++u) {
                    const float d   = dv[u];
                    const int   idx = chunk + nb + q * 4 + u;
                    if (d < kd[KNN_K - 1]) {   // rare path: fully predicated insert
#pragma unroll
                        for (int t = KNN_K - 1; t >= 0; --t) {
                            const bool gt    = kd[t] > d;
                            const bool shift = (t > 0) ? (kd[t - 1] > d) : false;
                            const float cd   = shift ? kd[t - 1] : d;
                            const int   cix  = shift ? ki[t - 1] : idx;
                            kd[t] = gt ? cd : kd[t];
                            ki[t] = gt ? cix : ki[t];
                        }
                    }
                }
            }
        }
        __syncthreads();        // compute done before next pack overwrites s_pts
    }

    // Finalize: add back |c|^2, clip, sqrt; write dist + idx
    const size_t obase = ((size_t)b * M + m_own) * KNN_K;
#pragma unroll
    for (int j = 0; j < KNN_K; ++j) {
        float dd = kd[j] + csq;
        dd = fmaxf(dd, KNN_EPS);
        out_dist[obase + j] = sqrtf(dd);
        out_idx [obase + j] = ki[j];
    }
}

extern "C" void kernel_launch(void* const* d_in, const int* in_sizes, int n_in,
                              void* d_out, int out_size, void* d_ws, size_t ws_size,
                              hipStream_t stream) {
    const float* coords  = (const float*)d_in[0];   // (B, N, 3) f32
    const float* centers = (const float*)d_in[1];   // (B, M, 3) f32

    const int B = 2, D = 3;
    const int N = in_sizes[0] / (B * D);            // 32768
    const int M = in_sizes[1] / (B * D);            // 8192

    float* out_dist = (float*)d_out;                                 // (B,M,K) f32
    int*   out_idx  = (int*)((float*)d_out + (size_t)B * M * KNN_K); // (B,M,K) i32

    dim3 grid(B * (M / ROWS_PER_BLOCK));            // 128 blocks
    knn_wmma_kernel<<<grid, BLOCK_THREADS, 0, stream>>>(coords, centers,
                                                        out_dist, out_idx, N, M);
}